// GraphFeatureExtractor_25340307046635
// MI455X (gfx1250) — compile-verified
//
#include <hip/hip_runtime.h>
#include <hip/hip_bf16.h>

// ---------------------------------------------------------------------------
// RGCN (2 layers) for MI455X / gfx1250, wave32 + WMMA f32 16x16x4.
//
//   H = x @ [W_0 .. W_{R-1} | root]           (dense, WMMA matrix pipe)
//   agg[dst] += H[src, etype-block] / indeg   (edge scatter, f32 atomics)
//   out = agg + root-part + b  (+ReLU layer1)
//
// Memory-bound op (~29 GFLOP vs ~1.5-2 GB traffic -> ~80us at 23.3 TB/s), so
// fp32 WMMA (exact reference numerics) is the right precision; no downcast.
// ---------------------------------------------------------------------------

typedef __attribute__((ext_vector_type(2))) float v2f;
typedef __attribute__((ext_vector_type(8))) float v8f;

#define N_NODES 100000
#define N_EDGES 600000
#define D_K     128
#define D_OUT1  128
#define D_OUT2  96
#define N_REL   4

// ---------------- zero-fill helpers ----------------------------------------
__global__ void zero_f32_kernel(float* __restrict__ p, int n) {
    int i = blockIdx.x * blockDim.x + threadIdx.x;
    if (i < n) p[i] = 0.0f;
}
__global__ void zero_i32_kernel(int* __restrict__ p, int n) {
    int i = blockIdx.x * blockDim.x + threadIdx.x;
    if (i < n) p[i] = 0;
}

// ---------------- per-(dst, etype) in-degree --------------------------------
__global__ void count_kernel(const int* __restrict__ dst,
                             const int* __restrict__ etype,
                             int* __restrict__ cnt, int nedges) {
    int e = blockIdx.x * blockDim.x + threadIdx.x;
    if (e >= nedges) return;
    atomicAdd(&cnt[dst[e] * N_REL + etype[e]], 1);
}

// ---------------- fused WMMA GEMM: H = A @ [W_0..W_{R-1} | root] ------------
// A: [nrows, 128] row-major.  W: [R, 128, DOUT] row-major.  root: [128, DOUT].
// H: [nrows, (R+1)*DOUT] row-major.
// Block: 256 threads = 8 waves. Block owns 16 rows; each wave owns one 16-col
// tile. A tile staged in LDS (stride 132 floats -> conflict-free for the
// half-lane K-pair access pattern of V_WMMA_F32_16X16X4_F32, and 16B-aligned
// rows for float4 staging). DOUT is compile-time so all B/H addressing is
// base + immediate offset (no per-element 64-bit address math).
#define LDS_STRIDE 132

template <int DOUT>
__global__ __launch_bounds__(256)
void gemm_wmma_f32_kernel(const float* __restrict__ A,
                          const float* __restrict__ W,
                          const float* __restrict__ root,
                          float* __restrict__ H) {
    constexpr int NCOLS = (N_REL + 1) * DOUT;
    __shared__ float ldsA[16 * LDS_STRIDE];

    const int rowBase = blockIdx.x * 16;   // N_NODES = 6250*16, exact

    // Cooperative float4 stage of the 16x128 A tile (2048 floats = 512 f4).
    {
        const float4* __restrict__ Ag = (const float4*)(A + (size_t)rowBase * D_K);
        for (int i = threadIdx.x; i < 16 * D_K / 4; i += 256) {
            const float4 v = Ag[i];
            const int r = (4 * i) >> 7;          // /128
            const int c = (4 * i) & (D_K - 1);   // %128, multiple of 4
            float* __restrict__ s = &ldsA[r * LDS_STRIDE + c];  // 16B aligned
            s[0] = v.x; s[1] = v.y; s[2] = v.z; s[3] = v.w;
        }
    }
    __syncthreads();

    const int wave = threadIdx.x >> 5;       // 0..7
    const int lane = threadIdx.x & 31;
    const int half = lane >> 4;              // 0: lanes 0-15, 1: lanes 16-31
    const int l16  = lane & 15;

    const int colTile = blockIdx.y * 8 + wave;      // wave-uniform
    if (colTile * 16 >= NCOLS) return;              // EXEC stays all-1s per wave
    const int colBase = colTile * 16;

    // Select B matrix (relation block or root block); both row-major [128,DOUT].
    const float* B;
    int bcol;
    if (colBase < N_REL * DOUT) {
        const int r = colBase / DOUT;
        B    = W + (size_t)r * D_K * DOUT;
        bcol = colBase - r * DOUT;
    } else {
        B    = root;
        bcol = colBase - N_REL * DOUT;
    }
    // Per-lane B base: row (2*half), column bcol+l16. K-loop offsets are
    // compile-time immediates from here on.
    const float* __restrict__ Bbase = B + 2 * half * DOUT + bcol + l16;
    const float* __restrict__ Arow  = &ldsA[l16 * LDS_STRIDE + 2 * half];

    v8f acc = {};
#pragma unroll
    for (int k = 0; k < D_K; k += 4) {
        // A fragment (16x4 f32): lanes 0-15 hold K=k,k+1; lanes 16-31 K=k+2,k+3.
        v2f a, b;
        a.x = Arow[k + 0];
        a.y = Arow[k + 1];
        // B fragment (4x16 f32): rows striped across lanes, K split across halves.
        b.x = Bbase[k * DOUT];
        b.y = Bbase[k * DOUT + DOUT];
        acc = __builtin_amdgcn_wmma_f32_16x16x4_f32(
                  /*neg_a=*/false, a, /*neg_b=*/false, b,
                  /*c_mod=*/(short)0, acc, /*reuse_a=*/false, /*reuse_b=*/false);
    }

    // C/D layout: VGPR v -> M = v + 8*half, N = l16.
    float* __restrict__ Hout =
        H + (size_t)(rowBase + half * 8) * NCOLS + colBase + l16;
#pragma unroll
    for (int v = 0; v < 8; ++v)
        Hout[v * NCOLS] = acc[v];
}

// ---------------- edge scatter: agg[dst] += H[src, etype-block] / indeg -----
template <int DOUT>
__global__ __launch_bounds__(256)
void scatter_kernel(const float* __restrict__ H,
                    const int* __restrict__ src,
                    const int* __restrict__ dst,
                    const int* __restrict__ etype,
                    const int* __restrict__ cnt,
                    float* __restrict__ agg,
                    int nedges) {
    constexpr int NCOLS = (N_REL + 1) * DOUT;
    const int wave = threadIdx.x >> 5;
    const int lane = threadIdx.x & 31;
    const int e = blockIdx.x * (blockDim.x >> 5) + wave;
    if (e >= nedges) return;

    const int s = src[e];
    const int d = dst[e];
    const int r = etype[e];
    const int c = cnt[d * N_REL + r];
    const float norm = 1.0f / fmaxf((float)c, 1.0f);

    const float* __restrict__ hrow = H + (size_t)s * NCOLS + r * DOUT;
    float* __restrict__ arow = agg + (size_t)d * DOUT;

#pragma unroll
    for (int j0 = 0; j0 < DOUT; j0 += 128) {
        const int j = j0 + lane * 4;
        if (j < DOUT) {
            const float4 m = *(const float4*)(hrow + j);
            atomicAdd(arow + j + 0, norm * m.x);
            atomicAdd(arow + j + 1, norm * m.y);
            atomicAdd(arow + j + 2, norm * m.z);
            atomicAdd(arow + j + 3, norm * m.w);
        }
    }
}

// ---------------- finalize: out = agg + root-part + bias (opt. ReLU) --------
template <int DOUT, bool RELU>
__global__ void finalize_kernel(const float* __restrict__ agg,
                                const float* __restrict__ Hcat,
                                const float* __restrict__ bias,
                                float* __restrict__ out,
                                int nrows) {
    constexpr int NCOLS = (N_REL + 1) * DOUT;
    const int i = blockIdx.x * blockDim.x + threadIdx.x;
    if (i >= nrows * DOUT) return;
    const int n = i / DOUT;          // compile-time divisor -> mul/shift
    const int d = i - n * DOUT;
    float v = agg[i] + Hcat[(size_t)n * NCOLS + N_REL * DOUT + d] + bias[d];
    if (RELU) v = fmaxf(v, 0.0f);
    out[i] = v;
}

// ---------------------------------------------------------------------------
extern "C" void kernel_launch(void* const* d_in, const int* in_sizes, int n_in,
                              void* d_out, int out_size, void* d_ws, size_t ws_size,
                              hipStream_t stream) {
    const float* x     = (const float*)d_in[0];   // [N, 128]
    const int*   ei    = (const int*)  d_in[1];   // [2, E]
    const int*   et    = (const int*)  d_in[2];   // [E]
    const float* W1    = (const float*)d_in[3];   // [R, 128, 128]
    const float* root1 = (const float*)d_in[4];   // [128, 128]
    const float* b1    = (const float*)d_in[5];   // [128]
    const float* W2    = (const float*)d_in[6];   // [R, 128, 96]
    const float* root2 = (const float*)d_in[7];   // [128, 96]
    const float* b2    = (const float*)d_in[8];   // [96]
    float* out = (float*)d_out;                   // [N, 96]

    const int* srcv = ei;
    const int* dstv = ei + N_EDGES;

    // Workspace layout.
    char* ws = (char*)d_ws;
    size_t off = 0;
    int* cnt = (int*)(ws + off);          off += (size_t)N_NODES * N_REL * 4;
    float* H1   = (float*)(ws + off);     off += (size_t)N_NODES * (N_REL + 1) * D_OUT1 * 4;
    float* agg1 = (float*)(ws + off);     off += (size_t)N_NODES * D_OUT1 * 4;
    float* h    = (float*)(ws + off);     off += (size_t)N_NODES * D_OUT1 * 4;
    float* H2   = (float*)(ws + off);     off += (size_t)N_NODES * (N_REL + 1) * D_OUT2 * 4;
    float* agg2 = (float*)(ws + off);     off += (size_t)N_NODES * D_OUT2 * 4;
    (void)off; (void)ws_size; (void)in_sizes; (void)n_in; (void)out_size;

    // 0) zero scratch
    zero_i32_kernel<<<(N_NODES * N_REL + 255) / 256, 256, 0, stream>>>(cnt, N_NODES * N_REL);
    zero_f32_kernel<<<(N_NODES * D_OUT1 + 255) / 256, 256, 0, stream>>>(agg1, N_NODES * D_OUT1);
    zero_f32_kernel<<<(N_NODES * D_OUT2 + 255) / 256, 256, 0, stream>>>(agg2, N_NODES * D_OUT2);

    // 1) in-degree per (dst, etype) — shared by both layers
    count_kernel<<<(N_EDGES + 255) / 256, 256, 0, stream>>>(dstv, et, cnt, N_EDGES);

    const int rowBlocks = N_NODES / 16;   // 6250, exact

    // 2) layer 1 dense: H1 = x @ [W1 | root1], 640 cols -> 40 tiles -> grid.y = 5
    {
        dim3 grid(rowBlocks, ((N_REL + 1) * D_OUT1 / 16 + 7) / 8);
        gemm_wmma_f32_kernel<D_OUT1><<<grid, 256, 0, stream>>>(x, W1, root1, H1);
    }
    // 3) layer 1 scatter
    scatter_kernel<D_OUT1><<<(N_EDGES + 7) / 8, 256, 0, stream>>>(
        H1, srcv, dstv, et, cnt, agg1, N_EDGES);
    // 4) layer 1 finalize + ReLU -> h
    finalize_kernel<D_OUT1, true><<<(N_NODES * D_OUT1 + 255) / 256, 256, 0, stream>>>(
        agg1, H1, b1, h, N_NODES);

    // 5) layer 2 dense: H2 = h @ [W2 | root2], 480 cols -> 30 tiles -> grid.y = 4
    {
        dim3 grid(rowBlocks, ((N_REL + 1) * D_OUT2 / 16 + 7) / 8);
        gemm_wmma_f32_kernel<D_OUT2><<<grid, 256, 0, stream>>>(h, W2, root2, H2);
    }
    // 6) layer 2 scatter
    scatter_kernel<D_OUT2><<<(N_EDGES + 7) / 8, 256, 0, stream>>>(
        H2, srcv, dstv, et, cnt, agg2, N_EDGES);
    // 7) layer 2 finalize -> out
    finalize_kernel<D_OUT2, false><<<(N_NODES * D_OUT2 + 255) / 256, 256, 0, stream>>>(
        agg2, H2, b2, out, N_NODES);
}